// LC1GNS_17901423690517
// MI455X (gfx1250) — compile-verified
//
#include <hip/hip_runtime.h>

#define WAVES 2
#define DT 0.01f

// ---- LDS weight layout offsets (floats) ----
#define ENC_C 0       // w1(16) b1(16) W2(256) b2(16)  = 304
#define ENC_L 304
#define ENC_I 608
#define ENC_N 912
#define PROC  1216    // P1(768) pb1(16) P2(256) pb2(16) = 1056
#define DEC_C 2272    // D1(256) db1(16) D2(256) db2(16) d3(16) db3(1) = 561
#define DEC_L 2833
#define DEC_N 3394
#define W_TOTAL 3955

typedef float v2f __attribute__((ext_vector_type(2)));
typedef float v8f __attribute__((ext_vector_type(8)));

struct GNParams { const float* p[38]; };

__device__ const int kSz[38] = {
    16,16,256,16,   // enc_C
    16,16,256,16,   // enc_L
    16,16,256,16,   // enc_I
    16,16,256,16,   // enc_node
    768,16,256,16,  // proc_node
    256,16,256,16,16,1,   // dec_C
    256,16,256,16,16,1,   // dec_L
    256,16,256,16,16,1    // dec_node
};

// Same-wave LDS ordering fence (CDNA5 split wait counters).
__device__ __forceinline__ void lds_fence() {
  asm volatile("s_wait_dscnt 0" ::: "memory");
}

__device__ __forceinline__ v8f bias_init(const float* b, int l) {
  float bv = b[l & 15];
  v8f acc;
#pragma unroll
  for (int r = 0; r < 8; ++r) acc[r] = bv;
  return acc;
}

__device__ __forceinline__ v8f zero_init() {
  v8f acc;
#pragma unroll
  for (int r = 0; r < 8; ++r) acc[r] = 0.f;
  return acc;
}

// acc += X(16x16 item-major) @ W (or W^T).  A: lanes0-15 K{0,1}, lanes16-31 K{2,3}.
template <bool TRANS>
__device__ __forceinline__ v8f mm16(const float* __restrict__ X,
                                    const float* __restrict__ W,
                                    v8f acc, int l) {
  const int n  = l & 15;       // A row (item) and B/D column coincide
  const int hi = l >> 4;
#pragma unroll
  for (int kc = 0; kc < 4; ++kc) {
    const int k0 = kc * 4 + 2 * hi;
    v2f a, b;
    a[0] = X[n * 16 + k0];
    a[1] = X[n * 16 + k0 + 1];
    if (TRANS) {               // B element (k,n) = W[n][k]
      b[0] = W[n * 16 + k0];
      b[1] = W[n * 16 + k0 + 1];
    } else {                   // B element (k,n) = W[k][n]
      b[0] = W[k0 * 16 + n];
      b[1] = W[(k0 + 1) * 16 + n];
    }
    acc = __builtin_amdgcn_wmma_f32_16x16x4_f32(false, a, false, b,
                                                (short)0, acc, false, false);
  }
  return acc;
}

// D layout: VGPR r -> row (r + 8*hi), col n.
__device__ __forceinline__ void tstore(float* Y, v8f acc, bool relu, int l) {
  const int n = l & 15; const int hi = l >> 4;
#pragma unroll
  for (int r = 0; r < 8; ++r) {
    float v = acc[r];
    if (relu) v = fmaxf(v, 0.f);
    Y[(r + 8 * hi) * 16 + n] = v;
  }
}

__device__ __forceinline__ void tstore_mask(float* Y, v8f acc, const float* M, int l) {
  const int n = l & 15; const int hi = l >> 4;
#pragma unroll
  for (int r = 0; r < 8; ++r) {
    const int row = r + 8 * hi;
    Y[row * 16 + n] = (M[row * 16 + n] > 0.f) ? acc[r] : 0.f;
  }
}

// First encoder layer: H[m][f] = relu(scale*xs[m*stride] * w1[f] + b1[f])
__device__ __forceinline__ void enc_l1(float* H, const float* xs, int stride,
                                       float scale, const float* w1,
                                       const float* b1, int l) {
#pragma unroll
  for (int r = 0; r < 8; ++r) {
    const int i = r * 32 + l;
    const int m = i >> 4, f = i & 15;
    const float x = scale * xs[m * stride];
    H[i] = fmaxf(x * w1[f] + b1[f], 0.f);
  }
}

// T[m][f] = (U[m][f] > 0) ? v[f] : 0
__device__ __forceinline__ void maskvec(float* T, const float* U, const float* v, int l) {
#pragma unroll
  for (int r = 0; r < 8; ++r) {
    const int i = r * 32 + l;
    T[i] = (U[i] > 0.f) ? v[i & 15] : 0.f;
  }
}

// res[m] = sum_f T[m][f]*v[f] + addc
__device__ __forceinline__ void rowdot(float* res, const float* T, const float* v,
                                       float addc, float* red, int l) {
  const int m = l & 15; const int hi = l >> 4;
  float s = 0.f;
#pragma unroll
  for (int j = 0; j < 8; ++j) {
    const int f = hi * 8 + j;
    s += T[m * 16 + f] * v[f];
  }
  red[l] = s;
  lds_fence();
  if (hi == 0) res[m] = red[m] + red[m + 16] + addc;
  lds_fence();
}

// Forward dec MLP (16->16->16->1) on EL, then backprop d(out)/d(enc input scalar).
// Forceinline so all tile pointers stay provably LDS (ds_* not flat_*).
__device__ __forceinline__ void dec_edge_path(
    const float* EL, const float* Henc, int dec_off, int enc_off,
    float* S0, float* S1, float* S2, float* S3,
    float* aOut, float* fOut, float* red, const float* sW, int l) {
  const float* D1  = sW + dec_off;
  const float* db1 = D1 + 256;
  const float* D2  = D1 + 272;
  const float* db2 = D1 + 528;
  const float* d3  = D1 + 544;
  const float  db3 = D1[560];
  const float* ew1 = sW + enc_off;
  const float* eW2 = sW + enc_off + 32;

  // forward
  v8f acc = bias_init(db1, l);
  acc = mm16<false>(EL, D1, acc, l);
  tstore(S0, acc, true, l); lds_fence();          // U1 (post-relu)
  acc = bias_init(db2, l);
  acc = mm16<false>(S0, D2, acc, l);
  tstore(S1, acc, true, l); lds_fence();          // U2 (post-relu)
  rowdot(aOut, S1, d3, db3, red, l);              // scalar dec output

  // backward
  maskvec(S2, S1, d3, l); lds_fence();            // t2 = d3 .* relu'(u2)
  acc = zero_init();
  acc = mm16<true>(S2, D2, acc, l);               // t2 @ D2^T
  tstore_mask(S3, acc, S0, l); lds_fence();       // t1
  acc = zero_init();
  acc = mm16<true>(S3, D1, acc, l);               // t1 @ D1^T
  tstore(S2, acc, false, l); lds_fence();         // grad wrt enc output y
  acc = zero_init();
  acc = mm16<true>(S2, eW2, acc, l);              // @ W2^T
  tstore_mask(S3, acc, Henc, l); lds_fence();     // t0 = .* relu'(h)
  rowdot(fOut, S3, ew1, 0.f, red, l);             // f'(x) = t0 . w1
}

// Full per-node path: enc_node -> proc (recv/sent via WMMA-accumulate) -> dec_node.
__device__ __forceinline__ void node_path(
    int i, const float* sNodes,
    const float* EL0, const float* EL1, const float* EL2, const float* EL3,
    float* S0, float* S1, float* S2,
    float* dnOut, float* red, const float* sW, int l) {
  const float* nw1 = sW + ENC_N; const float* nb1 = nw1 + 16;
  const float* nW2 = nw1 + 32;   const float* nb2 = nw1 + 288;
  const float* P1  = sW + PROC;  const float* pb1 = P1 + 768;
  const float* P2  = P1 + 784;   const float* pb2 = P1 + 1040;
  const float* dD1 = sW + DEC_N; const float* ddb1 = dD1 + 256;
  const float* dD2 = dD1 + 272;  const float* ddb2 = dD1 + 528;
  const float* dd3 = dD1 + 544;  const float  ddb3 = dD1[560];

  enc_l1(S0, sNodes + i, 3, 1.f, nw1, nb1, l); lds_fence();
  v8f acc = bias_init(nb2, l);
  acc = mm16<false>(S0, nW2, acc, l);
  tstore(S1, acc, false, l); lds_fence();          // node latent

  // proc layer 1: z = [nl, recv, sent] @ P1 as 3 K-blocks; segment sums folded
  // into extra accumulating WMMAs (distributivity).
  acc = bias_init(pb1, l);
  acc = mm16<false>(S1, P1, acc, l);               // nl block (rows 0-15)
  if (i == 0) {                                    // recv={}, sent={el0,el2}
    acc = mm16<false>(EL0, P1 + 512, acc, l);
    acc = mm16<false>(EL2, P1 + 512, acc, l);
  } else if (i == 1) {                             // recv={el0}, sent={el1}
    acc = mm16<false>(EL0, P1 + 256, acc, l);
    acc = mm16<false>(EL1, P1 + 512, acc, l);
  } else {                                         // recv={el1,el2,el3}, sent={el3}
    acc = mm16<false>(EL1, P1 + 256, acc, l);
    acc = mm16<false>(EL2, P1 + 256, acc, l);
    acc = mm16<false>(EL3, P1 + 256, acc, l);
    acc = mm16<false>(EL3, P1 + 512, acc, l);
  }
  tstore(S2, acc, true, l); lds_fence();
  acc = bias_init(pb2, l);
  acc = mm16<false>(S2, P2, acc, l);
  tstore(S0, acc, false, l); lds_fence();          // processed node latent

  // dec_node
  acc = bias_init(ddb1, l);
  acc = mm16<false>(S0, dD1, acc, l);
  tstore(S1, acc, true, l); lds_fence();
  acc = bias_init(ddb2, l);
  acc = mm16<false>(S1, dD2, acc, l);
  tstore(S2, acc, true, l); lds_fence();
  rowdot(dnOut, S2, dd3, ddb3, red, l);
}

__global__ __launch_bounds__(WAVES * 32)
void gns_kernel(const float* __restrict__ nodes, const float* __restrict__ edges,
                const float* __restrict__ control, float* __restrict__ out,
                GNParams P, int nItems) {
  __shared__ float sW[W_TOTAL];
  __shared__ float sT[WAVES][11][256];
  __shared__ float sS[WAVES][320];

  {
    int off = 0;
#pragma unroll
    for (int a = 0; a < 38; ++a) {
      const float* src = P.p[a];
      const int n = kSz[a];
      for (int i = threadIdx.x; i < n; i += blockDim.x) sW[off + i] = src[i];
      off += n;
    }
  }
  __syncthreads();

  const int wave = threadIdx.x >> 5;
  const int l    = threadIdx.x & 31;
  const int base = (blockIdx.x * WAVES + wave) * 16;
  if (base >= nItems) return;

  float* sNodes = sS[wave];
  float* sEdges = sNodes + 48;
  float* sCtrl  = sNodes + 112;
  float* red    = sNodes + 128;
  float* aArr   = sNodes + 160;  // 3 x 16 dec-edge scalars
  float* fArr   = sNodes + 208;  // 3 x 16 path derivatives
  float* dnArr  = sNodes + 256;  // 3 x 16 dec-node outputs

  for (int i = l; i < 48; i += 32) sNodes[i] = nodes[base * 3 + i];
  for (int i = l; i < 64; i += 32) sEdges[i] = edges[base * 4 + i];
  if (l < 16) sCtrl[l] = control[base + l];
  lds_fence();

  float* Hq  = sT[wave][0];
  float* Hmq = sT[wave][1];
  float* Hp  = sT[wave][2];
  float* EL0 = sT[wave][3];
  float* EL1 = sT[wave][4];
  float* EL2 = sT[wave][5];
  float* EL3 = sT[wave][6];
  float* S0  = sT[wave][7];
  float* S1  = sT[wave][8];
  float* S2  = sT[wave][9];
  float* S3  = sT[wave][10];

  const float* cw1 = sW + ENC_C; const float* cb1 = cw1 + 16;
  const float* cW2 = cw1 + 32;   const float* cb2 = cw1 + 288;
  const float* lw1 = sW + ENC_L; const float* lb1 = lw1 + 16;
  const float* lW2 = lw1 + 32;   const float* lb2 = lw1 + 288;
  const float* iw1 = sW + ENC_I; const float* ib1 = iw1 + 16;
  const float* iW2 = iw1 + 32;   const float* ib2 = iw1 + 288;

  // edge encoders: el0=encC(q), el2=encC(-q), el1=encL(p), el3=encI(c)
  enc_l1(Hq,  sEdges + 0, 4,  1.f, cw1, cb1, l);
  enc_l1(Hmq, sEdges + 0, 4, -1.f, cw1, cb1, l);
  enc_l1(Hp,  sEdges + 1, 4,  1.f, lw1, lb1, l);
  enc_l1(S0,  sEdges + 3, 4,  1.f, iw1, ib1, l);
  lds_fence();
  v8f acc;
  acc = bias_init(cb2, l); acc = mm16<false>(Hq,  cW2, acc, l); tstore(EL0, acc, false, l);
  acc = bias_init(cb2, l); acc = mm16<false>(Hmq, cW2, acc, l); tstore(EL2, acc, false, l);
  acc = bias_init(lb2, l); acc = mm16<false>(Hp,  lW2, acc, l); tstore(EL1, acc, false, l);
  acc = bias_init(ib2, l); acc = mm16<false>(S0,  iW2, acc, l); tstore(EL3, acc, false, l);
  lds_fence();

  // node pipeline (enc_node -> proc -> dec_node), fully specialized per node
#pragma unroll
  for (int i = 0; i < 3; ++i)
    node_path(i, sNodes, EL0, EL1, EL2, EL3, S0, S1, S2, dnArr + 16 * i, red, sW, l);

  // H terms + gradients (dec_I never reaches the output -> skipped)
  dec_edge_path(EL0, Hq,  DEC_C, ENC_C, S0, S1, S2, S3, aArr +  0, fArr +  0, red, sW, l);
  dec_edge_path(EL1, Hp,  DEC_L, ENC_L, S0, S1, S2, S3, aArr + 16, fArr + 16, red, sW, l);
  dec_edge_path(EL2, Hmq, DEC_C, ENC_C, S0, S1, S2, S3, aArr + 32, fArr + 32, red, sW, l);

  // outputs: [dn0,dn1,dn2, nq, np, -nq, control, H] per item
  for (int i = l; i < 128; i += 32) {
    const int m = i >> 3, c = i & 7;
    const float q  = sEdges[4 * m];
    const float p  = sEdges[4 * m + 1];
    const float cc = sEdges[4 * m + 3];
    const float H    = aArr[m] + aArr[16 + m] + aArr[32 + m];
    const float dHdq = fArr[m] - fArr[32 + m];   // el2 = encC(-q): chain rule sign
    const float dHdp = fArr[16 + m];
    const float nq = q + DT * (dHdp + cc);
    const float np = p - DT * dHdq;
    float v;
    switch (c) {
      case 0: v = dnArr[m];      break;
      case 1: v = dnArr[16 + m]; break;
      case 2: v = dnArr[32 + m]; break;
      case 3: v = nq;            break;
      case 4: v = np;            break;
      case 5: v = -nq;           break;
      case 6: v = sCtrl[m];      break;
      default: v = H;            break;
    }
    out[(base + m) * 8 + c] = v;
  }
}

extern "C" void kernel_launch(void* const* d_in, const int* in_sizes, int n_in,
                              void* d_out, int out_size, void* d_ws, size_t ws_size,
                              hipStream_t stream) {
  (void)d_ws; (void)ws_size; (void)out_size;
  if (n_in < 47) return;
  const float* nodes   = (const float*)d_in[0];
  const float* edges   = (const float*)d_in[1];
  const float* control = (const float*)d_in[2];
  const int B = in_sizes[2];

  // params flattened in sorted-dict-key pytree order:
  // dec_C:3-8 dec_I:9-14 dec_L:15-20 dec_node:21-26
  // enc_C:27-30 enc_I:31-34 enc_L:35-38 enc_node:39-42 proc_node:43-46
  GNParams P; int k = 0;
  for (int i = 0; i < 4; ++i) P.p[k++] = (const float*)d_in[27 + i]; // enc_C
  for (int i = 0; i < 4; ++i) P.p[k++] = (const float*)d_in[35 + i]; // enc_L
  for (int i = 0; i < 4; ++i) P.p[k++] = (const float*)d_in[31 + i]; // enc_I
  for (int i = 0; i < 4; ++i) P.p[k++] = (const float*)d_in[39 + i]; // enc_node
  for (int i = 0; i < 4; ++i) P.p[k++] = (const float*)d_in[43 + i]; // proc_node
  for (int i = 0; i < 6; ++i) P.p[k++] = (const float*)d_in[3 + i];  // dec_C
  for (int i = 0; i < 6; ++i) P.p[k++] = (const float*)d_in[15 + i]; // dec_L
  for (int i = 0; i < 6; ++i) P.p[k++] = (const float*)d_in[21 + i]; // dec_node

  const int tiles  = (B + 15) / 16;
  const int blocks = (tiles + WAVES - 1) / WAVES;
  gns_kernel<<<blocks, WAVES * 32, 0, stream>>>(nodes, edges, control,
                                                (float*)d_out, P, B);
}